// LSTNet_29068338660009
// MI455X (gfx1250) — compile-verified
//
#include <hip/hip_runtime.h>
#include <math.h>

// ---------------- constants ----------------
#define LL 24      // history length
#define PP 12      // horizon
#define EPSBN 1e-5f
#define SLOPE 0.01f
#define BDIM 256

typedef float v2f __attribute__((ext_vector_type(2)));
typedef float v8f __attribute__((ext_vector_type(8)));

// Param block layout (36 floats, padded to 64):
// 0 wc_last, 1 bc_last, 2 wc_cur, 3 bc_cur, 4 g_st, 5 b_st,
// 6-8 Wih, 9-11 Whh, 12-14 bih, 15-17 bhh,
// 18 g_gru, 19 b_gru, 20 wl, 21 wr, 22-34 comb_w[13], 35 comb_b

__global__ void gather_params(float* __restrict__ dst,
    const float* wc_last, const float* bc_last, const float* wc_cur, const float* bc_cur,
    const float* g_st, const float* b_st,
    const float* Wih, const float* Whh, const float* bih, const float* bhh,
    const float* g_gru, const float* b_gru, const float* wl, const float* wr,
    const float* comb_w, const float* comb_b)
{
    if (threadIdx.x == 0) {
        dst[0] = wc_last[0]; dst[1] = bc_last[0]; dst[2] = wc_cur[0]; dst[3] = bc_cur[0];
        dst[4] = g_st[0];    dst[5] = b_st[0];
        for (int i = 0; i < 3; ++i) {
            dst[6+i]  = Wih[i]; dst[9+i]  = Whh[i];
            dst[12+i] = bih[i]; dst[15+i] = bhh[i];
        }
        dst[18] = g_gru[0]; dst[19] = b_gru[0]; dst[20] = wl[0]; dst[21] = wr[0];
        for (int i = 0; i < 13; ++i) dst[22+i] = comb_w[i];
        dst[35] = comb_b[0];
    }
}

__device__ __forceinline__ void block_reduce_pair(float s, float q, float* __restrict__ part) {
    __shared__ float sh_s[BDIM];
    __shared__ float sh_q[BDIM];
    int tid = threadIdx.x;
    sh_s[tid] = s; sh_q[tid] = q;
    __syncthreads();
    #pragma unroll
    for (int w = BDIM / 2; w > 0; w >>= 1) {
        if (tid < w) { sh_s[tid] += sh_s[tid + w]; sh_q[tid] += sh_q[tid + w]; }
        __syncthreads();
    }
    if (tid == 0) { part[2 * blockIdx.x] = sh_s[0]; part[2 * blockIdx.x + 1] = sh_q[0]; }
}

// Pass 1: global sum/sumsq of st = wc_last*x[t-1] + wc_cur*x[t] + bc  (x[-1]==0)
__global__ void stats_st(const float* __restrict__ X, const float* __restrict__ Pb,
                         float* __restrict__ part, int M)
{
    int m = blockIdx.x * BDIM + threadIdx.x;
    const float wcl = Pb[0], wcc = Pb[2], bc = Pb[1] + Pb[3];
    float s = 0.f, q = 0.f;
    if (m < M) {
        float prev = 0.f;
        #pragma unroll 4
        for (int t = 0; t < LL; ++t) {
            float cur = X[(size_t)t * M + m];
            float st  = fmaf(wcl, prev, fmaf(wcc, cur, bc));
            s += st; q += st * st;
            prev = cur;
        }
    }
    block_reduce_pair(s, q, part);
}

// Final reduce of per-block partials -> [mean, rsqrt(var+eps)]
__global__ void reduce_stats(const float* __restrict__ part, int n, float count,
                             float* __restrict__ out2)
{
    float s = 0.f, q = 0.f;
    for (int i = threadIdx.x; i < n; i += BDIM) { s += part[2*i]; q += part[2*i+1]; }
    __shared__ float sh_s[BDIM];
    __shared__ float sh_q[BDIM];
    int tid = threadIdx.x;
    sh_s[tid] = s; sh_q[tid] = q;
    __syncthreads();
    #pragma unroll
    for (int w = BDIM / 2; w > 0; w >>= 1) {
        if (tid < w) { sh_s[tid] += sh_s[tid + w]; sh_q[tid] += sh_q[tid + w]; }
        __syncthreads();
    }
    if (tid == 0) {
        float mean = sh_s[0] / count;
        float var  = sh_q[0] / count - mean * mean;   // biased, matches torch/jax norm
        out2[0] = mean;
        out2[1] = rsqrtf(var + EPSBN);
    }
}

__device__ __forceinline__ float fast_sigmoid(float x) {
    return 1.f / (1.f + __expf(-x));
}
__device__ __forceinline__ float fast_tanh(float x) {
    x = fminf(fmaxf(x, -15.f), 15.f);
    float e = __expf(2.f * x);
    return (e - 1.f) / (e + 1.f);
}

// Pass 2: normalized st -> scalar GRU scan; emit last PP outputs to o[PP][M],
// and accumulate sum/sumsq of those outputs for the second BatchNorm.
__global__ void gru_scan(const float* __restrict__ X, const float* __restrict__ Pb,
                         const float* __restrict__ st1, float* __restrict__ o,
                         float* __restrict__ part, int M)
{
    int m = blockIdx.x * BDIM + threadIdx.x;
    const float wcl = Pb[0], wcc = Pb[2], bc = Pb[1] + Pb[3];
    const float g = Pb[4], b = Pb[5];
    const float Wih0 = Pb[6],  Wih1 = Pb[7],  Wih2 = Pb[8];
    const float Whh0 = Pb[9],  Whh1 = Pb[10], Whh2 = Pb[11];
    const float bih0 = Pb[12], bih1 = Pb[13], bih2 = Pb[14];
    const float bhh0 = Pb[15], bhh1 = Pb[16], bhh2 = Pb[17];
    const float mean = st1[0], inv = st1[1];
    float s = 0.f, q = 0.f;
    if (m < M) {
        float h = 0.f, prev = 0.f;
        #pragma unroll 4
        for (int t = 0; t < LL; ++t) {
            float cur = X[(size_t)t * M + m];
            if (t + 2 < LL) __builtin_prefetch(&X[(size_t)(t + 2) * M + m], 0, 3);
            float st = fmaf(wcl, prev, fmaf(wcc, cur, bc));
            prev = cur;
            float x  = fmaf(g * inv, st - mean, b);
            float r  = fast_sigmoid(fmaf(Wih0, x, bih0) + fmaf(Whh0, h, bhh0));
            float z  = fast_sigmoid(fmaf(Wih1, x, bih1) + fmaf(Whh1, h, bhh1));
            float nn = fast_tanh(fmaf(Wih2, x, bih2) + r * fmaf(Whh2, h, bhh2));
            h = (1.f - z) * nn + z * h;
            if (t >= LL - PP) {
                o[(size_t)(t - (LL - PP)) * M + m] = h;
                s += h; q += h * h;
            }
        }
    }
    block_reduce_pair(s, q, part);
}

// stack column k for cell m (after 2nd BN): k<12 -> alpha_k * o_k ; k==12 -> last ; else 0
__device__ __forceinline__ float stack_val(const float* __restrict__ o, int M, int m, int k,
                                           float last, float g, float b, float wl, float wr,
                                           float mean, float inv)
{
    if (k > 12) return 0.f;
    if (k == 12) return last;
    float on    = fmaf(g * inv, o[(size_t)k * M + m] - mean, b);
    float pre   = fmaf(wl, on, wr * last);
    float alpha = pre > 0.f ? pre : SLOPE * pre;
    return alpha * on;
}

// Pass 3: per-16-cell tile, out = stack(16x16-K-padded) @ comb_w via V_WMMA_F32_16X16X4_F32.
// B is comb_w broadcast across all 16 N columns, so every column of D equals the result.
// A layout (32-bit 16x4): lanes 0-15 hold K={0,1}, lanes 16-31 hold K={2,3} per chunk.
__global__ void combine_wmma(const float* __restrict__ o, const float* __restrict__ Pb,
                             const float* __restrict__ st2, const float* __restrict__ resAR,
                             float* __restrict__ out, int M)
{
    const int lane = threadIdx.x & 31;
    const int wave = (blockIdx.x * blockDim.x + threadIdx.x) >> 5;
    const int tiles = M >> 4;
    if (wave >= tiles) return;                 // wave-uniform; EXEC stays all-1s

    const int m0 = wave << 4;
    const int r  = lane & 15;
    const int hi = lane >> 4;
    const int m  = m0 + r;

    const float g = Pb[18], b = Pb[19], wl = Pb[20], wr = Pb[21], cb = Pb[35];
    const float mean = st2[0], inv = st2[1];

    const float last = fmaf(g * inv, o[(size_t)(PP - 1) * M + m] - mean, b);

    v8f acc = {0.f, 0.f, 0.f, 0.f, 0.f, 0.f, 0.f, 0.f};
    #pragma unroll
    for (int c = 0; c < 4; ++c) {
        const int k0 = 4 * c + 2 * hi;
        const int k1 = k0 + 1;
        v2f a, bv;
        a.x  = stack_val(o, M, m, k0, last, g, b, wl, wr, mean, inv);
        a.y  = stack_val(o, M, m, k1, last, g, b, wl, wr, mean, inv);
        bv.x = (k0 < 13) ? Pb[22 + k0] : 0.f;
        bv.y = (k1 < 13) ? Pb[22 + k1] : 0.f;
        acc = __builtin_amdgcn_wmma_f32_16x16x4_f32(
            /*neg_a=*/false, a, /*neg_b=*/false, bv,
            /*c_mod=*/(short)0, acc, /*reuse_a=*/false, /*reuse_b=*/false);
    }

    // D layout: VGPR v holds row v (lanes 0-15) / row v+8 (lanes 16-31); all N cols equal.
    if (r == 0) {
        const int base = m0 + hi * 8;
        #pragma unroll
        for (int v = 0; v < 8; ++v)
            out[base + v] = acc[v] + cb + resAR[base + v];
    }
}

// ---------------- host-side launch ----------------
extern "C" void kernel_launch(void* const* d_in, const int* in_sizes, int n_in,
                              void* d_out, int out_size, void* d_ws, size_t ws_size,
                              hipStream_t stream)
{
    const float* Ds   = (const float*)d_in[0];   // [24,16,256]
    const float* Gs   = (const float*)d_in[1];   // [24,16,256,256]
    const float* resD = (const float*)d_in[2];   // [16,256]
    const float* resG = (const float*)d_in[3];   // [16,256,256]

    const int MD = 16 * 256;          // 4096 cells
    const int MG = 16 * 256 * 256;    // 1,048,576 cells

    float* ws    = (float*)d_ws;
    float* pD    = ws;                // 64
    float* pG    = ws + 64;           // 64
    float* stD   = ws + 128;          // mean1,inv1,mean2,inv2
    float* stG   = ws + 160;
    float* partD = ws + 1024;         // 16 blocks * 2
    float* partG = ws + 2048;         // 4096 blocks * 2 = 8192
    float* oD    = ws + 20480;        // 12*4096  = 49152
    float* oG    = ws + 131072;       // 12*1048576 = 12582912 (~51 MB total ws use)

    float* outD = (float*)d_out;
    float* outG = (float*)d_out + MD;

    // ---- gather scalar params into contiguous blocks ----
    gather_params<<<1, 64, 0, stream>>>(pD,
        (const float*)d_in[4],  (const float*)d_in[5],  (const float*)d_in[6],  (const float*)d_in[7],
        (const float*)d_in[8],  (const float*)d_in[9],  (const float*)d_in[10], (const float*)d_in[11],
        (const float*)d_in[12], (const float*)d_in[13], (const float*)d_in[14], (const float*)d_in[15],
        (const float*)d_in[16], (const float*)d_in[17], (const float*)d_in[18], (const float*)d_in[19]);
    gather_params<<<1, 64, 0, stream>>>(pG,
        (const float*)d_in[20], (const float*)d_in[21], (const float*)d_in[22], (const float*)d_in[23],
        (const float*)d_in[24], (const float*)d_in[25], (const float*)d_in[26], (const float*)d_in[27],
        (const float*)d_in[28], (const float*)d_in[29], (const float*)d_in[30], (const float*)d_in[31],
        (const float*)d_in[32], (const float*)d_in[33], (const float*)d_in[34], (const float*)d_in[35]);

    // ---- D branch ----
    {
        const int nb = MD / BDIM;    // 16
        stats_st<<<nb, BDIM, 0, stream>>>(Ds, pD, partD, MD);
        reduce_stats<<<1, BDIM, 0, stream>>>(partD, nb, (float)(LL * (long)MD), stD);
        gru_scan<<<nb, BDIM, 0, stream>>>(Ds, pD, stD, oD, partD, MD);
        reduce_stats<<<1, BDIM, 0, stream>>>(partD, nb, (float)(PP * (long)MD), stD + 2);
        combine_wmma<<<(MD / 16) / 8, BDIM, 0, stream>>>(oD, pD, stD + 2, resD, outD, MD);
    }
    // ---- G branch ----
    {
        const int nb = MG / BDIM;    // 4096
        stats_st<<<nb, BDIM, 0, stream>>>(Gs, pG, partG, MG);
        reduce_stats<<<1, BDIM, 0, stream>>>(partG, nb, (float)(LL * (long)MG), stG);
        gru_scan<<<nb, BDIM, 0, stream>>>(Gs, pG, stG, oG, partG, MG);
        reduce_stats<<<1, BDIM, 0, stream>>>(partG, nb, (float)(PP * (long)MG), stG + 2);
        combine_wmma<<<(MG / 16) / 8, BDIM, 0, stream>>>(oG, pG, stG + 2, resG, outG, MG);
    }
}